// MedianPool2d_5128190951772
// MI455X (gfx1250) — compile-verified
//
#include <hip/hip_runtime.h>
#include <hip/hip_bf16.h>
#include <stdint.h>

// 3x3 median pool, stride 1, reflect pad 1, on float32 (8,8,512,512).
// Bandwidth-bound: 128 MiB min traffic -> ~5.6us floor at 23.3 TB/s.
// Data path: global -> LDS via gfx1250 async loads (ASYNCcnt), median-of-9
// min/max network in registers, 128-bit non-temporal stores.

typedef float v4f __attribute__((ext_vector_type(4)));
typedef float v2f __attribute__((ext_vector_type(2)));

#define HW 512
#define TX 128                 // tile width  (outputs)
#define TY 16                  // tile height (outputs)
#define LDSW 132               // (TX+2) halo, padded to mult-of-4 floats -> 16B-aligned rows
#define LDSH (TY + 2)
#define NFILL (LDSH * LDSW)    // 2376 elements
#define NTHREADS 256

__device__ __forceinline__ int refl(int i, int n) {
  // reflect (no edge repeat): -1 -> 1, n -> n-2 ; valid for i in [-n, 2n-2)
  i = (i < 0) ? -i : i;
  return (i >= n) ? (2 * n - 2 - i) : i;
}

__device__ __forceinline__ void s2(float& a, float& b) {
  float t = fminf(a, b);
  b = fmaxf(a, b);
  a = t;
}

// classic 19-compare-exchange median-of-9 network; returns exact median
__device__ __forceinline__ float med9(float p0, float p1, float p2,
                                      float p3, float p4, float p5,
                                      float p6, float p7, float p8) {
  s2(p1, p2); s2(p4, p5); s2(p7, p8);
  s2(p0, p1); s2(p3, p4); s2(p6, p7);
  s2(p1, p2); s2(p4, p5); s2(p7, p8);
  s2(p0, p3); s2(p5, p8); s2(p4, p7);
  s2(p3, p6); s2(p1, p4); s2(p2, p5);
  s2(p4, p7); s2(p4, p2); s2(p6, p4);
  s2(p4, p2);
  return p4;
}

__global__ __launch_bounds__(NTHREADS)
void median3x3_kernel(const float* __restrict__ x, float* __restrict__ out) {
  __shared__ float tile[NFILL];   // 9504 B of the 320 KB WGP LDS

  const int tid = threadIdx.x;
  const int x0 = blockIdx.x * TX;
  const int y0 = blockIdx.y * TY;
  const size_t plane_off = (size_t)blockIdx.z * (size_t)(HW * HW);
  const float* src = x + plane_off;

  // ---- async fill: global -> LDS, reflect-padded halo tile ------------------
  // Per-lane reflected global address, per-lane LDS byte offset; data moves
  // memory->LDS on the async path (no VGPR round trip), tracked by ASYNCcnt.
  for (int idx = tid; idx < NFILL; idx += NTHREADS) {
    int ti = idx / LDSW;
    int tj = idx - ti * LDSW;
    int gy = refl(y0 - 1 + ti, HW);
    int gx = refl(x0 - 1 + tj, HW);
    const float* gp = src + gy * HW + gx;
    unsigned lds_off = (unsigned)(uintptr_t)(tile + idx);  // low 32 bits = LDS offset
    asm volatile("global_load_async_to_lds_b32 %0, %1, off"
                 :: "v"(lds_off), "v"(gp) : "memory");
  }
  asm volatile("s_wait_asynccnt 0x0" ::: "memory");
  __syncthreads();

  // ---- compute: each thread -> two float4 output rows -----------------------
  const int c4  = tid & 31;     // 32 column-quads across the tile
  const int r   = tid >> 5;     // 8 row slots; handles rows r and r+8
  const int col = c4 * 4;

  float* dst = out + plane_off;

#pragma unroll
  for (int k = 0; k < 2; ++k) {
    const int rr = r + k * 8;   // output row within tile (0..15)
    float f[3][6];              // 3 halo rows x 6 columns (covers 4 windows)
#pragma unroll
    for (int dr = 0; dr < 3; ++dr) {
      const float* base = tile + (rr + dr) * LDSW + col;  // 16B aligned
      v4f a = *(const v4f*)base;          // ds_load_b128
      v2f b = *(const v2f*)(base + 4);    // ds_load_b64
      f[dr][0] = a.x; f[dr][1] = a.y; f[dr][2] = a.z; f[dr][3] = a.w;
      f[dr][4] = b.x; f[dr][5] = b.y;
    }
    v4f o;
#pragma unroll
    for (int j = 0; j < 4; ++j) {
      o[j] = med9(f[0][j], f[0][j + 1], f[0][j + 2],
                  f[1][j], f[1][j + 1], f[1][j + 2],
                  f[2][j], f[2][j + 1], f[2][j + 2]);
    }
    // Output is write-once, never re-read: non-temporal 128-bit store keeps
    // L2 free for input-halo reuse between neighboring tiles.
    size_t off = (size_t)(y0 + rr) * HW + (size_t)(x0 + col);
    __builtin_nontemporal_store(o, (v4f*)(dst + off));
  }
}

extern "C" void kernel_launch(void* const* d_in, const int* in_sizes, int n_in,
                              void* d_out, int out_size, void* d_ws, size_t ws_size,
                              hipStream_t stream) {
  const float* x = (const float*)d_in[0];
  float* out = (float*)d_out;
  // 64 (B*C) planes, each tiled 4 x 32 into 128x16 output tiles
  dim3 grid(HW / TX, HW / TY, 64);
  dim3 block(NTHREADS);
  median3x3_kernel<<<grid, block, 0, stream>>>(x, out);
}